// LogReg_455266533602
// MI455X (gfx1250) — compile-verified
//
#include <hip/hip_runtime.h>
#include <hip/hip_bf16.h>

// out[b] = bias + sum_{t<seqLen} W[text[t,b]]   (histogram @ W collapses to gather-sum)
//
// CDNA5 strategy:
//  * TDM (tensor_load_to_lds) DMAs the whole 128,000B W table into LDS once per block
//    (wave 0 issues; s_wait_tensorcnt 0; barrier). Gathers then come from 64-bank LDS.
//  * text (int64, values < 32768) is read as low dwords, coalesced 16-lane rows.
//  * The length-seqLen reduction runs on V_WMMA_F32_16X16X4_F32 with an all-ones B:
//    D[m][n] = sum_k A[m][k], accumulated in f32 C across seqLen/4 steps.

typedef __attribute__((ext_vector_type(2))) float v2f;
typedef __attribute__((ext_vector_type(8))) float v8f;
typedef __attribute__((ext_vector_type(4))) unsigned int v4u;
typedef __attribute__((ext_vector_type(8))) int v8i;
typedef __attribute__((ext_vector_type(4))) int v4i;

__global__ __launch_bounds__(512) void logreg_gather_wmma(
    const int* __restrict__ textLo,   // int64 tokens viewed as dword pairs (low dword used)
    const float* __restrict__ W,      // [vocab]
    const float* __restrict__ bias,   // [1]
    float* __restrict__ out,          // [batch]
    int batch, int seqLen, int vocab)
{
    extern __shared__ float sW[];     // vocab * 4 bytes of dynamic LDS

    // ---- Stage W into LDS via the Tensor Data Mover (wave 0 only) ----
    if (threadIdx.x < 32) {
        unsigned long long ga = (unsigned long long)W;
        // D# group 0: count=1, lds_addr=0, global_addr, type=2 ("image")
        v4u g0;
        g0.x = 1u;                                             // count=1, user descriptor
        g0.y = 0u;                                             // lds_addr (dynamic LDS base)
        g0.z = (unsigned)(ga & 0xFFFFFFFFull);                 // global_addr[31:0]
        g0.w = (unsigned)((ga >> 32) & 0x01FFFFFFull) | (2u << 30); // addr[56:32] | type=2
        // D# group 1: data_size=4B, tensor_dim0=vocab, tensor_dim1=1,
        //             tile_dim0=vocab, tile_dim1=1, tensor_dim0_stride=vocab
        unsigned uv = (unsigned)vocab;
        v8i g1;
        g1[0] = (int)(2u << 16);                               // wg_mask=0, data_size=2 (4B)
        g1[1] = (int)((uv & 0xFFFFu) << 16);                   // tensor_dim0[15:0] @ [31:16]
        g1[2] = (int)((uv >> 16) | (1u << 16));                // tensor_dim0[31:16] | tensor_dim1[15:0]=1
        g1[3] = (int)((uv & 0xFFFFu) << 16);                   // tensor_dim1[31:16]=0 | tile_dim0=vocab
        g1[4] = 1;                                             // tile_dim1=1, tile_dim2=0
        g1[5] = (int)uv;                                       // tensor_dim0_stride[31:0]
        g1[6] = 0;                                             // stride hi / dim1_stride lo
        g1[7] = 0;
        v4i g2 = {0, 0, 0, 0};
        v4i g3 = {0, 0, 0, 0};
        v8i g4 = {0, 0, 0, 0, 0, 0, 0, 0};                     // clang-23 6-arg form: extra group
        __builtin_amdgcn_tensor_load_to_lds(g0, g1, g2, g3, g4, 0);
        __builtin_amdgcn_s_wait_tensorcnt(0);
    }
    __syncthreads();

    const float bv   = bias[0];
    const int lane   = threadIdx.x & 31;
    const int wave   = threadIdx.x >> 5;
    const int base   = (blockIdx.x * (blockDim.x >> 5) + wave) * 16;  // 16 phrases per wave
    const int m      = lane & 15;            // row of A this lane feeds
    const int kb     = (lane >> 4) << 1;     // K base: lanes 0-15 -> {0,1}, lanes 16-31 -> {2,3}
    int p = base + m;
    if (p >= batch) p = batch - 1;           // clamp (keeps EXEC all-ones for WMMA)

    v8f c = {};                              // f32 accumulator (C/D)
    v2f ones; ones.x = 1.0f; ones.y = 1.0f;  // B = all-ones 4x16

    const int steps = seqLen >> 2;           // seqLen assumed multiple of 4 (100)
    for (int s = 0; s < steps; ++s) {
        const int t0 = (s << 2) + kb;
        const int f0 = t0 * batch + p;       // flat token index, row t0
        const int f1 = f0 + batch;           // row t0+1
        v2f a;
        a.x = sW[textLo[2 * f0]];            // low dword of int64 token id
        a.y = sW[textLo[2 * f1]];
        // D = A(16x4) * ones(4x16) + C  => every column of D holds the 4-token partial sum
        c = __builtin_amdgcn_wmma_f32_16x16x4_f32(
                /*neg_a=*/false, a, /*neg_b=*/false, ones,
                /*c_mod=*/(short)0, c, /*reuse_a=*/false, /*reuse_b=*/false);
    }

    // C/D layout: VGPR j, lanes 0-15 -> M=j (N=lane); lanes 16-31 -> M=j+8.
    // All N columns are identical (B was all-ones), so lane 0 / lane 16 suffice.
    if (lane == 0) {
        #pragma unroll
        for (int j = 0; j < 8; ++j) {
            int q = base + j;
            if (q < batch) out[q] = c[j] + bv;
        }
    } else if (lane == 16) {
        #pragma unroll
        for (int j = 0; j < 8; ++j) {
            int q = base + 8 + j;
            if (q < batch) out[q] = c[j] + bv;
        }
    }
}

extern "C" void kernel_launch(void* const* d_in, const int* in_sizes, int n_in,
                              void* d_out, int out_size, void* d_ws, size_t ws_size,
                              hipStream_t stream) {
    const int*   textLo = (const int*)d_in[0];    // int64 [seq_len, batch] -> dword pairs
    const float* W      = (const float*)d_in[1];  // [1, vocab]
    const float* bptr   = (const float*)d_in[2];  // [1]
    float*       out    = (float*)d_out;          // [batch, 1]

    const int batch  = out_size;                  // 8192
    const int seqLen = in_sizes[0] / batch;       // 100
    const int vocab  = in_sizes[1];               // 32000

    const int threads = 512;                      // 16 waves
    const int phrasesPerBlock = (threads / 32) * 16;  // 256
    const int grid = (batch + phrasesPerBlock - 1) / phrasesPerBlock;
    const size_t smem = (size_t)vocab * sizeof(float);  // 128,000 B of LDS (<= 320KB/WGP)

    logreg_gather_wmma<<<grid, threads, smem, stream>>>(
        textLo, W, bptr, out, batch, seqLen, vocab);
}